// BiLSTM_57191784513898
// MI455X (gfx1250) — compile-verified
//
#include <hip/hip_runtime.h>
#include <math.h>

#define T_LEN 1024
#define B_SZ  64
#define D_SZ  256
#define H_SZ  256
#define KTOT  512           // D + H
#define G4H   1024          // 4*H
#define NWG_DIR 16
#define WAVES_PER_WG 4
#define BLOCK_THREADS 128
#define BSTRIDE 516         // padded per-column LDS stride in floats (bank-conflict free)
#define LDS_BYTES (64 * BSTRIDE * 4)

typedef float v2f __attribute__((ext_vector_type(2)));
typedef float v8f __attribute__((ext_vector_type(8)));

__device__ __forceinline__ float sigmoidf_(float v) {
    return 1.0f / (1.0f + expf(-v));
}

// -------- init kernel: zero barrier counters + h double-buffers -----------
extern "C" __global__ void __launch_bounds__(256)
bilstm_init(unsigned int* ws_u32, int nwords) {
    int i = blockIdx.x * blockDim.x + threadIdx.x;
    if (i < nwords) ws_u32[i] = 0u;
}

// -------- persistent fused BiLSTM scan kernel -----------------------------
// grid: 2 * NWG_DIR blocks of 128 threads (4 waves).
// Each wave owns one 16x16 (b-tile, h-tile) of the c/h state for one direction.
extern "C" __global__ void __launch_bounds__(BLOCK_THREADS)
bilstm_scan(const float* __restrict__ x,
            const float* __restrict__ fwd_W, const float* __restrict__ fwd_U,
            const float* __restrict__ fwd_b,
            const float* __restrict__ bwd_W, const float* __restrict__ bwd_U,
            const float* __restrict__ bwd_b,
            float* __restrict__ out,               // (T,B,2H) ++ h_f(B,H) ++ h_b(B,H)
            unsigned int* __restrict__ bar_base,   // ws: [dir*32]
            float* __restrict__ hbuf_base)         // ws+256B: [dir][2][B*H]
{
    const int dir  = blockIdx.x / NWG_DIR;
    const int wgd  = blockIdx.x % NWG_DIR;         // == h-tile index (0..15)
    const int wave = threadIdx.x >> 5;             // == b-tile index (0..3)
    const int lane = threadIdx.x & 31;

    const int b0 = wave * 16;                      // batch-row base of this wave's tile
    const int n0 = wgd * 16;                       // h-column base of this WG

    const float* W    = dir ? bwd_W : fwd_W;
    const float* U    = dir ? bwd_U : fwd_U;
    const float* bias = dir ? bwd_b : fwd_b;
    unsigned int* bar = bar_base + dir * 32;
    float* hbuf = hbuf_base + (size_t)dir * 2 * (B_SZ * H_SZ);

    // ---- stage this WG's weight slice into LDS once (reused 1024 steps) ----
    // local col c in [0,64): gate g = c>>4, global z-column = g*H + n0 + (c&15)
    extern __shared__ float Blds[];                // [64][BSTRIDE]
    for (int idx = threadIdx.x; idx < 64 * KTOT; idx += BLOCK_THREADS) {
        int c  = idx >> 9;                         // / 512
        int k  = idx & (KTOT - 1);
        int gN = (c >> 4) * H_SZ + n0 + (c & 15);
        float v = (k < D_SZ) ? W[(size_t)k * G4H + gN]
                             : U[(size_t)(k - D_SZ) * G4H + gN];
        Blds[c * BSTRIDE + k] = v;
    }
    __syncthreads();

    // ---- per-lane fragment coordinates (ISA 7.12.2 layouts) ----
    const int m   = lane & 15;                     // A row / B,C,D column within tile
    const int kc2 = (lane >> 4) << 1;              // k sub-offset {0,2}
    const int chi = (lane >> 4) * 8;               // C/D row-half offset
    const int r   = b0 + m;                        // batch row read by this lane for A

    const float bf_ = bias[0 * H_SZ + n0 + m];
    const float bg_ = bias[1 * H_SZ + n0 + m];
    const float bi_ = bias[2 * H_SZ + n0 + m];
    const float bo_ = bias[3 * H_SZ + n0 + m];

    v8f cstate = {};                               // c-tile lives in VGPRs

    for (int t = 0; t < T_LEN; ++t) {
        const int tt = dir ? (T_LEN - 1 - t) : t;  // actual time index
        const float* h_read  = hbuf + (size_t)(t & 1) * (B_SZ * H_SZ);
        float*       h_write = hbuf + (size_t)((t & 1) ^ 1) * (B_SZ * H_SZ);

        const float* xrow = x + ((size_t)tt * B_SZ + r) * D_SZ;
        const float* hrow = h_read + (size_t)r * H_SZ;

        v8f zf = {}, zg = {}, zi = {}, zo = {};

        // K = 0..255 : x_t @ W part
        #pragma unroll 4
        for (int kk = 0; kk < D_SZ; kk += 4) {
            v2f a  = *(const v2f*)(xrow + kk + kc2);
            const float* bp = Blds + kk + kc2;
            v2f bF = *(const v2f*)(bp + (0 * 16 + m) * BSTRIDE);
            v2f bG = *(const v2f*)(bp + (1 * 16 + m) * BSTRIDE);
            v2f bI = *(const v2f*)(bp + (2 * 16 + m) * BSTRIDE);
            v2f bO = *(const v2f*)(bp + (3 * 16 + m) * BSTRIDE);
            zf = __builtin_amdgcn_wmma_f32_16x16x4_f32(false, a, false, bF, (short)0, zf, false, false);
            zg = __builtin_amdgcn_wmma_f32_16x16x4_f32(false, a, false, bG, (short)0, zg, false, false);
            zi = __builtin_amdgcn_wmma_f32_16x16x4_f32(false, a, false, bI, (short)0, zi, false, false);
            zo = __builtin_amdgcn_wmma_f32_16x16x4_f32(false, a, false, bO, (short)0, zo, false, false);
        }
        // K = 256..511 : h_{t-1} @ U part
        #pragma unroll 4
        for (int kk = 0; kk < H_SZ; kk += 4) {
            v2f a  = *(const v2f*)(hrow + kk + kc2);
            const float* bp = Blds + (D_SZ + kk) + kc2;
            v2f bF = *(const v2f*)(bp + (0 * 16 + m) * BSTRIDE);
            v2f bG = *(const v2f*)(bp + (1 * 16 + m) * BSTRIDE);
            v2f bI = *(const v2f*)(bp + (2 * 16 + m) * BSTRIDE);
            v2f bO = *(const v2f*)(bp + (3 * 16 + m) * BSTRIDE);
            zf = __builtin_amdgcn_wmma_f32_16x16x4_f32(false, a, false, bF, (short)0, zf, false, false);
            zg = __builtin_amdgcn_wmma_f32_16x16x4_f32(false, a, false, bG, (short)0, zg, false, false);
            zi = __builtin_amdgcn_wmma_f32_16x16x4_f32(false, a, false, bI, (short)0, zi, false, false);
            zo = __builtin_amdgcn_wmma_f32_16x16x4_f32(false, a, false, bO, (short)0, zo, false, false);
        }

        // ---- gates + state update (c stays in registers) ----
        v8f hval;
        #pragma unroll
        for (int j = 0; j < 8; ++j) {
            float f = sigmoidf_(zf[j] + bf_);
            float g = tanhf(zg[j] + bg_);
            float i = sigmoidf_(zi[j] + bi_);
            float o = sigmoidf_(zo[j] + bo_);
            float c = cstate[j] * f + g * i;
            cstate[j] = c;
            hval[j]   = o * tanhf(c);
        }

        // ---- write h tile: hs output + next-step h buffer ----
        #pragma unroll
        for (int j = 0; j < 8; ++j) {
            int brow = b0 + chi + j;
            int col  = n0 + m;
            float hv = hval[j];
            h_write[(size_t)brow * H_SZ + col] = hv;
            out[((size_t)tt * B_SZ + brow) * (2 * H_SZ) + dir * H_SZ + col] = hv;
            if (t == T_LEN - 1) {   // final hidden state (h_f / h_b)
                out[(size_t)T_LEN * B_SZ * 2 * H_SZ + (size_t)dir * (B_SZ * H_SZ)
                    + (size_t)brow * H_SZ + col] = hv;
            }
        }

        // ---- per-direction grid barrier (monotonic counter, no reset races) ----
        __syncthreads();
        __threadfence();
        if (threadIdx.x == 0) {
            __hip_atomic_fetch_add(bar, 1u, __ATOMIC_RELEASE, __HIP_MEMORY_SCOPE_AGENT);
            const unsigned int target = (unsigned int)(t + 1) * NWG_DIR;
            while (__hip_atomic_load(bar, __ATOMIC_ACQUIRE, __HIP_MEMORY_SCOPE_AGENT) < target) {
                __builtin_amdgcn_s_sleep(1);
            }
        }
        __syncthreads();
        __threadfence();   // invalidate caches so fresh h is visible to all waves
    }
}

extern "C" void kernel_launch(void* const* d_in, const int* in_sizes, int n_in,
                              void* d_out, int out_size, void* d_ws, size_t ws_size,
                              hipStream_t stream) {
    const float* x     = (const float*)d_in[0];
    const float* fwd_W = (const float*)d_in[1];
    const float* fwd_U = (const float*)d_in[2];
    const float* fwd_b = (const float*)d_in[3];
    const float* bwd_W = (const float*)d_in[4];
    const float* bwd_U = (const float*)d_in[5];
    const float* bwd_b = (const float*)d_in[6];
    float* out = (float*)d_out;

    unsigned int* ws_u32 = (unsigned int*)d_ws;
    float* hbuf = (float*)((char*)d_ws + 256);

    // counters (64 words incl. padding) + 2 dirs * 2 buffers * B*H floats
    const int init_words = 64 + 2 * 2 * B_SZ * H_SZ;
    bilstm_init<<<(init_words + 255) / 256, 256, 0, stream>>>(ws_u32, init_words);

    (void)hipFuncSetAttribute(reinterpret_cast<const void*>(bilstm_scan),
                              hipFuncAttributeMaxDynamicSharedMemorySize, LDS_BYTES);

    bilstm_scan<<<2 * NWG_DIR, BLOCK_THREADS, LDS_BYTES, stream>>>(
        x, fwd_W, fwd_U, fwd_b, bwd_W, bwd_U, bwd_b, out, ws_u32, hbuf);
}